// AdaptiveBoxBlurNd_74311524155671
// MI455X (gfx1250) — compile-verified
//
#include <hip/hip_runtime.h>
#include <stdint.h>

#define HDIM 768
#define WDIM 768
#define BDIM 4
#define CDIM 16
#define HW   (HDIM*WDIM)
#define NIMG (BDIM*CDIM)
#define NPART 72
#define EPSF 1e-5f

__device__ __forceinline__ int iclampi(int v, int lo, int hi){ return v<lo?lo:(v>hi?hi:v); }

__device__ __forceinline__ float reflectf(float v, float span){
  v = fabsf(v);
  float extra = fmodf(v, span);
  float flips = floorf(v / span);
  float r = (fmodf(flips, 2.0f) == 0.0f) ? extra : (span - extra);
  return fminf(fmaxf(r, 0.0f), span);
}

// ---------------- Kernel 1a: per-channel partial sum / sumsq ----------------
__global__ void __launch_bounds__(256) k1_partial(const float* __restrict__ x,
                                                  float* __restrict__ part)
{
  int t = threadIdx.x;
  int c = blockIdx.x / NPART;
  int p = blockIdx.x % NPART;
  int b = p / 18;          // HW = 18 * 32768 exactly
  int chunk = p % 18;
  size_t base = (size_t)(b*CDIM + c)*HW + (size_t)chunk*32768;
  float s = 0.f, q = 0.f;
  #pragma unroll 4
  for (int i = 0; i < 128; ++i){
    float v = x[base + (size_t)i*256 + t];
    s += v; q += v*v;
  }
  __shared__ float s1[256], s2[256];
  s1[t] = s; s2[t] = q; __syncthreads();
  for (int st = 128; st > 0; st >>= 1){
    if (t < st){ s1[t] += s1[t+st]; s2[t] += s2[t+st]; }
    __syncthreads();
  }
  if (t == 0){ part[2*blockIdx.x] = s1[0]; part[2*blockIdx.x+1] = s2[0]; }
}

// ---------------- Kernel 1b: finalize mean/std per channel ----------------
__global__ void __launch_bounds__(128) k1_finish(const float* __restrict__ part,
                                                 float* __restrict__ stats)
{
  int c = blockIdx.x;
  int t = threadIdx.x;
  float s = 0.f, q = 0.f;
  if (t < NPART){ s = part[2*(c*NPART+t)]; q = part[2*(c*NPART+t)+1]; }
  __shared__ float s1[128], s2[128];
  s1[t] = s; s2[t] = q; __syncthreads();
  for (int st = 64; st > 0; st >>= 1){
    if (t < st){ s1[t] += s1[t+st]; s2[t] += s2[t+st]; }
    __syncthreads();
  }
  if (t == 0){
    const float N = (float)((size_t)BDIM*HW);
    float mean = s1[0] / N;
    float var  = (s2[0] - s1[0]*s1[0]/N) / (N - 1.0f);   // ddof=1
    stats[c]        = mean;
    stats[CDIM + c] = sqrtf(fmaxf(var, 0.f));
  }
}

// ---------------- Kernel 2: normalize + row prefix sum (wave per row) ------
// CDNA5 path: async global->LDS copy + s_wait_asynccnt
__global__ void __launch_bounds__(256) k2_rowscan(const float* __restrict__ x,
                                                  const float* __restrict__ stats,
                                                  float* __restrict__ sat)
{
  __shared__ float lds[8*WDIM];
  unsigned lane = threadIdx.x & 31u;
  unsigned wid  = (unsigned)__builtin_amdgcn_readfirstlane((int)(threadIdx.x >> 5));
  size_t row = (size_t)blockIdx.x*8 + wid;                 // global row id over all 64 images
  const float* rowp = x + row*WDIM;
  unsigned lbase = (unsigned)(uintptr_t)&lds[wid*WDIM];    // low 32 bits == LDS byte offset
  #pragma unroll
  for (int i = 0; i < 6; ++i){
    unsigned off = (unsigned)i*512u + lane*16u;            // 6 x (32 lanes x 16B) = 3072B row
    asm volatile("global_load_async_to_lds_b128 %0, %1, %2"
                 :: "v"(lbase + off), "v"(off), "s"(rowp) : "memory");
  }
  int c = (int)((row / HDIM) % CDIM);
  float mean = stats[c];
  float inv  = 1.0f / (stats[CDIM + c] + EPSF);
  asm volatile("s_wait_asynccnt 0" ::: "memory");

  const float4* lp = (const float4*)&lds[wid*WDIM + lane*24];
  float v[24];
  #pragma unroll
  for (int i = 0; i < 6; ++i){
    float4 qq = lp[i];
    v[4*i+0]=qq.x; v[4*i+1]=qq.y; v[4*i+2]=qq.z; v[4*i+3]=qq.w;
  }
  float run = 0.f;
  #pragma unroll
  for (int j = 0; j < 24; ++j){ run += (v[j]-mean)*inv; v[j] = run; }
  // wave-level exclusive scan of per-lane chunk totals (wave32)
  float sc = run;
  #pragma unroll
  for (int d = 1; d < 32; d <<= 1){
    float o = __shfl_up(sc, (unsigned)d, 32);
    if ((int)lane >= d) sc += o;
  }
  float excl = sc - run;
  float4* op = (float4*)(sat + row*WDIM + lane*24);
  #pragma unroll
  for (int i = 0; i < 6; ++i){
    float4 qq;
    qq.x = v[4*i+0]+excl; qq.y = v[4*i+1]+excl;
    qq.z = v[4*i+2]+excl; qq.w = v[4*i+3]+excl;
    op[i] = qq;
  }
}

// ---------------- Kernel 3: column running sum (thread per column) ---------
__global__ void __launch_bounds__(256) k3_colscan(float* __restrict__ sat)
{
  int img = blockIdx.x / 3;
  int seg = blockIdx.x % 3;
  int col = seg*256 + threadIdx.x;
  float* p = sat + (size_t)img*HW + col;
  float run = 0.f;
  for (int y = 0; y < HDIM; y += 8){
    if (y + 24 < HDIM) __builtin_prefetch(p + (size_t)(y+24)*WDIM, 1, 1);
    #pragma unroll
    for (int k = 0; k < 8; ++k){
      float v = p[(size_t)(y+k)*WDIM];
      run += v;
      p[(size_t)(y+k)*WDIM] = run;
    }
  }
}

// ---------------- Kernel 4: 4-corner bilinear SAT gather + denorm ----------
__global__ void __launch_bounds__(256) k4_gather(const float* __restrict__ sat,
                                                 const float* __restrict__ ks,
                                                 const float* __restrict__ stats,
                                                 float* __restrict__ out)
{
  __shared__ float sm[CDIM], sd[CDIM];
  int t = threadIdx.x;
  if (t < CDIM){ sm[t] = stats[t]; sd[t] = stats[CDIM + t]; }
  __syncthreads();

  int idx = blockIdx.x*256 + t;                // (b,y,x) flat, exactly B*H*W
  int b   = idx / HW;
  int rem = idx - b*HW;
  int y   = rem / WDIM;
  int x   = rem - y*WDIM;

  float kx = ks[(size_t)idx*2 + 0];
  float ky = ks[(size_t)idx*2 + 1];

  const float span = 767.0f;
  float gx = (float)x * (2.0f/767.0f) - 1.0f - (1.0f/768.0f);
  float gy = (float)y * (2.0f/767.0f) - 1.0f - (1.0f/768.0f);
  float fxk = kx * (1.0f/768.0f);
  float fyk = ky * (1.0f/768.0f);

  int   offs[16];
  float wts[16];
  const float cxs[4] = {-1.f,-1.f, 1.f, 1.f};
  const float cys[4] = {-1.f, 1.f,-1.f, 1.f};
  const float sgn[4] = { 1.f,-1.f,-1.f, 1.f};
  #pragma unroll
  for (int cr = 0; cr < 4; ++cr){
    float fx = gx + cxs[cr]*fxk;
    float fy = gy + cys[cr]*fyk;
    float ix = reflectf((fx + 1.0f)*0.5f*span, span);
    float iy = reflectf((fy + 1.0f)*0.5f*span, span);
    float x0f = floorf(ix), y0f = floorf(iy);
    float wx = ix - x0f,   wy = iy - y0f;
    int x0 = iclampi((int)x0f, 0, WDIM-1);
    int x1 = iclampi(x0 + 1,  0, WDIM-1);
    int y0 = iclampi((int)y0f, 0, HDIM-1);
    int y1 = iclampi(y0 + 1,  0, HDIM-1);
    float s = sgn[cr];
    offs[cr*4+0] = y0*WDIM + x0; wts[cr*4+0] = s*(1.f-wx)*(1.f-wy);
    offs[cr*4+1] = y0*WDIM + x1; wts[cr*4+1] = s*wx*(1.f-wy);
    offs[cr*4+2] = y1*WDIM + x0; wts[cr*4+2] = s*(1.f-wx)*wy;
    offs[cr*4+3] = y1*WDIM + x1; wts[cr*4+3] = s*wx*wy;
  }

  float rarea = 1.0f / (kx*ky + EPSF);
  const float* base = sat + (size_t)b*CDIM*HW;
  float* ob = out + (size_t)b*CDIM*HW + rem;
  for (int c = 0; c < CDIM; ++c){
    const float* img = base + (size_t)c*HW;
    float s = 0.f;
    #pragma unroll
    for (int q = 0; q < 16; ++q) s += wts[q]*img[offs[q]];
    ob[(size_t)c*HW] = s * rarea * sd[c] + sm[c];
  }
}

// ---------------------------------------------------------------------------
extern "C" void kernel_launch(void* const* d_in, const int* in_sizes, int n_in,
                              void* d_out, int out_size, void* d_ws, size_t ws_size,
                              hipStream_t stream)
{
  (void)in_sizes; (void)n_in; (void)out_size; (void)ws_size;
  const float* x  = (const float*)d_in[0];   // [4,16,768,768]
  const float* ks = (const float*)d_in[1];   // [4,768,768,2]
  float* out = (float*)d_out;

  float* part  = (float*)d_ws;                          // 2*16*72 floats
  float* stats = (float*)((char*)d_ws + 16384);         // mean[16], std[16]
  float* sat   = (float*)((char*)d_ws + 65536);         // 64 * 768*768 fp32 SAT (151 MB)

  hipLaunchKernelGGL(k1_partial, dim3(CDIM*NPART), dim3(256), 0, stream, x, part);
  hipLaunchKernelGGL(k1_finish,  dim3(CDIM),       dim3(128), 0, stream, part, stats);
  hipLaunchKernelGGL(k2_rowscan, dim3(NIMG*HDIM/8), dim3(256), 0, stream, x, stats, sat);
  hipLaunchKernelGGL(k3_colscan, dim3(NIMG*3),     dim3(256), 0, stream, sat);
  hipLaunchKernelGGL(k4_gather,  dim3(BDIM*HW/256), dim3(256), 0, stream, sat, ks, stats, out);
}